// Normal_NoLayer_11141145166392
// MI455X (gfx1250) — compile-verified
//
#include <hip/hip_runtime.h>

// dims from the reference
#define DB    2
#define DN    1024
#define DS    64      // seq_in * nh_in = 16*4 neighbors
#define DM    128     // nv * nc = 4*32 output rows per window
#define DXY   16      // nlon * nlat
#define DNC   32
#define MIN_SIGMA 1e-10f

typedef __attribute__((ext_vector_type(2))) float v2f;
typedef __attribute__((ext_vector_type(8))) float v8f;

// One workgroup (8 wave32) per (b,n) window.
//  wave w computes C rows m in [16w, 16w+16), all 16 xy columns:
//  C[128x16] = A[128x64] * B[64x16], A = x^T, B = normalized Gaussian weights.
__global__ __launch_bounds__(256)
void gauss_basis_agg_kernel(const float* __restrict__ x,
                            const float* __restrict__ clon,
                            const float* __restrict__ clat,
                            const float* __restrict__ mus_lon,
                            const float* __restrict__ mus_lat,
                            const float* __restrict__ sigma,
                            float* __restrict__ out)
{
    __shared__ float s_w[DS][DXY];       // normalized weight matrix, 64x16 f32
    __shared__ float s_inv[DXY];         // 1 / sum_s w[s][xy]
    __shared__ float s_out[4][DXY][36];  // staging for coalesced stores
                                         // (stride 36: 16B-aligned rows, 16 banks)

    const int win  = blockIdx.x;          // 0 .. B*N-1
    const int tid  = threadIdx.x;         // 0 .. 255
    const int lane = tid & 31;
    const int wave = tid >> 5;

    const float sig     = fmaxf(sigma[0], MIN_SIGMA);
    const float inv_sig = 1.0f / sig;

    // ---- Phase 1: Gaussian weights into LDS -------------------------------
    const float* cl = clon + (size_t)win * DS;   // coords flat: s = si*4 + nh
    const float* ct = clat + (size_t)win * DS;
    #pragma unroll
    for (int i = tid; i < DS * DXY; i += 256) {
        const int s  = i >> 4;
        const int xy = i & 15;
        const int lo = xy >> 2;
        const int la = xy & 3;
        const float dlo = (cl[s] - mus_lon[lo]) * inv_sig;
        const float dla = (ct[s] - mus_lat[la]) * inv_sig;
        s_w[s][xy] = __expf(-0.5f * (dlo * dlo + dla * dla));
    }
    __syncthreads();

    if (tid < DXY) {
        float sum = 0.0f;
        #pragma unroll
        for (int s = 0; s < DS; ++s) sum += s_w[s][tid];
        s_inv[tid] = 1.0f / sum;
    }
    __syncthreads();

    // Fold the neighbor-axis normalization into the weights once per block,
    // so B fragments feed WMMA straight from ds_load.
    #pragma unroll
    for (int i = tid; i < DS * DXY; i += 256) {
        s_w[i >> 4][i & 15] *= s_inv[i & 15];
    }
    __syncthreads();

    // ---- Phase 2: 16 chained V_WMMA_F32_16X16X4_F32 -----------------------
    // A 16x4 f32 layout: lane L -> row M = L%16; VGPR j holds K = 2*(L/16)+j.
    // B 4x16 f32 layout: lane L -> col N = L%16; VGPR j holds K = 2*(L/16)+j.
    const float* xw  = x + (size_t)win * (DS * DM);   // x[s][m], m = v*32+c
    const int mrow   = wave * 16 + (lane & 15);       // output row this lane feeds
    const int khalf  = lane >> 4;                     // 0: K pair {0,1}, 1: {2,3}
    const int xy     = lane & 15;

    v8f acc = {};
    #pragma unroll
    for (int kk = 0; kk < DS / 4; ++kk) {
        const int sb = kk * 4 + 2 * khalf;
        v2f a, b;
        a.x = xw[(size_t)(sb    ) * DM + mrow];
        a.y = xw[(size_t)(sb + 1) * DM + mrow];
        b.x = s_w[sb    ][xy];
        b.y = s_w[sb + 1][xy];
        acc = __builtin_amdgcn_wmma_f32_16x16x4_f32(
                  /*neg_a=*/false, a, /*neg_b=*/false, b,
                  /*c_mod=*/(short)0, acc,
                  /*reuse_a=*/false, /*reuse_b=*/false);
    }

    // ---- Phase 3: stage C fragments in LDS, then coalesced b128 stores ----
    // C 16x16 f32 layout: VGPR r, lanes 0-15 -> M=r, lanes 16-31 -> M=r+8,
    // N = lane%16.  Final order: idx = v*512 + xy*32 + c, with m = v*32 + c.
    #pragma unroll
    for (int r = 0; r < 8; ++r) {
        const int M = wave * 16 + r + 8 * khalf;
        s_out[M >> 5][xy][M & 31] = acc[r];
    }
    __syncthreads();

    float* ow = out + (size_t)win * (4 * DXY * DNC);  // 2048 floats per window
    {
        const int i0  = tid * 8;          // 8 consecutive output floats/thread
        const int v   = i0 >> 9;
        const int rem = i0 & 511;
        const int oxy = rem >> 5;
        const int c0  = rem & 31;         // multiple of 8 -> 16B aligned in LDS
        const float4 lo = *(const float4*)&s_out[v][oxy][c0];
        const float4 hi = *(const float4*)&s_out[v][oxy][c0 + 4];
        float4* dst = (float4*)(ow + i0);
        dst[0] = lo;
        dst[1] = hi;
    }
}

extern "C" void kernel_launch(void* const* d_in, const int* in_sizes, int n_in,
                              void* d_out, int out_size, void* d_ws, size_t ws_size,
                              hipStream_t stream) {
    const float* x        = (const float*)d_in[0];
    const float* clon     = (const float*)d_in[1];
    const float* clat     = (const float*)d_in[2];
    const float* mus_lon  = (const float*)d_in[3];
    const float* mus_lat  = (const float*)d_in[4];
    const float* sigma    = (const float*)d_in[5];
    float* out            = (float*)d_out;

    dim3 grid(DB * DN);   // 2048 windows, one workgroup each
    dim3 block(256);      // 8 wave32
    gauss_basis_agg_kernel<<<grid, block, 0, stream>>>(
        x, clon, clat, mus_lon, mus_lat, sigma, out);
}